// myGATConv_semi_13975823582068
// MI455X (gfx1250) — compile-verified
//
#include <hip/hip_runtime.h>
#include <hip/hip_bf16.h>

typedef __attribute__((ext_vector_type(2))) float v2f;
typedef __attribute__((ext_vector_type(8))) float v8f;

#define N_NODES   100000
#define N_EDGES   1600000
#define IN_FEATS  128
#define EDGE_FEATS 64
#define N_HEADS   8
#define OUT_FEATS 16
#define HD        (N_HEADS * OUT_FEATS)   // 128
#define EH        ((size_t)N_EDGES * N_HEADS)
#define NH        ((size_t)N_NODES * N_HEADS)
#define ND        ((size_t)N_NODES * HD)

// ---------------------------------------------------------------------------
// Kernel 0: init accumulators (agg=0 in d_out, e_max=-inf, denom=0, asum=0)
// ---------------------------------------------------------------------------
__global__ void gat_init_kernel(float* __restrict__ agg,
                                float* __restrict__ emax,
                                float* __restrict__ denom,
                                float* __restrict__ asum) {
    size_t i = (size_t)blockIdx.x * blockDim.x + threadIdx.x;
    size_t stride = (size_t)gridDim.x * blockDim.x;
    for (size_t j = i; j < ND; j += stride) agg[j] = 0.0f;
    for (size_t j = i; j < NH; j += stride) {
        emax[j]  = -__builtin_inff();
        denom[j] = 0.0f;
        asum[j]  = 0.0f;
    }
}

// ---------------------------------------------------------------------------
// Kernel 1: feat_proj = feat @ W   [100000x128] x [128x128]
// One wave computes a 16-row strip across all 128 output cols (8 N-tiles).
// f32 WMMA 16x16x4, K looped 128/4 = 32 steps. 100000 rows = 6250 tiles exact.
// ---------------------------------------------------------------------------
__global__ __launch_bounds__(256) void gat_gemm_node_kernel(
        const float* __restrict__ feat, const float* __restrict__ W,
        float* __restrict__ out) {
    const int lane  = threadIdx.x;                 // 0..31 (wave32)
    const int wave  = threadIdx.y;                 // 0..7
    const int tile  = blockIdx.x * 8 + wave;       // 16-row tile index
    if (tile >= N_NODES / 16) return;              // wave-uniform (EXEC all-1)
    const int m0    = tile * 16;
    const int mA    = lane & 15;
    const int kOff  = (lane >> 4) * 2;             // 0 or 2

    v8f acc[8];
#pragma unroll
    for (int t = 0; t < 8; ++t) acc[t] = {};

    for (int k = 0; k < IN_FEATS; k += 4) {
        // A fragment: A[mA][k+kOff], A[mA][k+kOff+1]  (8B aligned)
        const float* arow = feat + (size_t)(m0 + mA) * IN_FEATS + k + kOff;
        v2f afrag = *(const v2f*)arow;
#pragma unroll
        for (int t = 0; t < 8; ++t) {
            const int n0 = t * 16;
            // B fragment: B[k+kOff][n0+mA], B[k+kOff+1][n0+mA]
            const float* brow = W + (size_t)(k + kOff) * HD + n0 + mA;
            v2f bfrag;
            bfrag.x = brow[0];
            bfrag.y = brow[HD];
            acc[t] = __builtin_amdgcn_wmma_f32_16x16x4_f32(
                false, afrag, false, bfrag, (short)0, acc[t], false, false);
        }
    }

    const int mD = (lane < 16) ? 0 : 8;
#pragma unroll
    for (int t = 0; t < 8; ++t) {
        const int n0 = t * 16;
#pragma unroll
        for (int j = 0; j < 8; ++j) {
            out[(size_t)(m0 + j + mD) * HD + n0 + mA] = acc[t][j];
        }
    }
}

// ---------------------------------------------------------------------------
// Kernel 2: e = e_feat @ W_e  [1.6Mx64] x [64x8], N padded 8->16 in LDS.
// One wave per 16-edge tile, K=64 in 16 WMMA steps. 1.6M/16 = 100000 exact.
// ---------------------------------------------------------------------------
__global__ __launch_bounds__(256) void gat_gemm_edge_kernel(
        const float* __restrict__ efeat, const float* __restrict__ We,
        float* __restrict__ elogit) {
    __shared__ float sW[EDGE_FEATS * 16];
    const int tid = threadIdx.y * 32 + threadIdx.x;
    for (int i = tid; i < EDGE_FEATS * 16; i += 256) {
        int kk = i >> 4, nn = i & 15;
        sW[i] = (nn < N_HEADS) ? We[kk * N_HEADS + nn] : 0.0f;
    }
    __syncthreads();

    const int lane = threadIdx.x;
    const int tile = blockIdx.x * 8 + threadIdx.y;
    if (tile >= N_EDGES / 16) return;              // wave-uniform
    const int e0   = tile * 16;
    const int mA   = lane & 15;
    const int kOff = (lane >> 4) * 2;

    v8f acc = {};
    for (int k = 0; k < EDGE_FEATS; k += 4) {
        const float* arow = efeat + (size_t)(e0 + mA) * EDGE_FEATS + k + kOff;
        v2f afrag = *(const v2f*)arow;
        v2f bfrag;
        bfrag.x = sW[(k + kOff) * 16 + mA];
        bfrag.y = sW[(k + kOff + 1) * 16 + mA];
        acc = __builtin_amdgcn_wmma_f32_16x16x4_f32(
            false, afrag, false, bfrag, (short)0, acc, false, false);
    }

    const int mD = (lane < 16) ? 0 : 8;
    if (mA < N_HEADS) {                            // only first 8 cols are real
#pragma unroll
        for (int j = 0; j < 8; ++j) {
            elogit[(size_t)(e0 + j + mD) * N_HEADS + mA] = acc[j];
        }
    }
}

// ---------------------------------------------------------------------------
// Kernel 3: segment max by dst (float atomic max via ordered int bits)
// ---------------------------------------------------------------------------
__device__ __forceinline__ void atomicMaxFloat(float* addr, float v) {
    if (v >= 0.0f) atomicMax((int*)addr, __float_as_int(v));
    else           atomicMin((unsigned int*)addr, __float_as_uint(v));
}

__global__ void gat_segmax_kernel(const float* __restrict__ e,
                                  const int* __restrict__ dst,
                                  float* __restrict__ emax) {
    size_t i = (size_t)blockIdx.x * blockDim.x + threadIdx.x;
    if (i >= EH) return;
    int edge = (int)(i >> 3), h = (int)(i & 7);
    int d = dst[edge];
    atomicMaxFloat(emax + (size_t)d * N_HEADS + h, e[i]);
}

// ---------------------------------------------------------------------------
// Kernel 4: ex = exp(e - emax[dst]); denom += ex  (ex overwrites e in ws)
// ---------------------------------------------------------------------------
__global__ void gat_exp_kernel(float* __restrict__ e,
                               const int* __restrict__ dst,
                               const float* __restrict__ emax,
                               float* __restrict__ denom) {
    size_t i = (size_t)blockIdx.x * blockDim.x + threadIdx.x;
    if (i >= EH) return;
    int edge = (int)(i >> 3), h = (int)(i & 7);
    int d = dst[edge];
    float ex = expf(e[i] - emax[(size_t)d * N_HEADS + h]);
    e[i] = ex;
    atomicAdd(denom + (size_t)d * N_HEADS + h, ex);
}

// ---------------------------------------------------------------------------
// Kernel 5: a = ex / denom[dst]; attn_sum += a   (a -> d_out second region)
// ---------------------------------------------------------------------------
__global__ void gat_attn_kernel(const float* __restrict__ ex,
                                const int* __restrict__ dst,
                                const float* __restrict__ denom,
                                float* __restrict__ a_out,
                                float* __restrict__ asum) {
    size_t i = (size_t)blockIdx.x * blockDim.x + threadIdx.x;
    if (i >= EH) return;
    int edge = (int)(i >> 3), h = (int)(i & 7);
    int d = dst[edge];
    float a = ex[i] / denom[(size_t)d * N_HEADS + h];
    a_out[i] = a;
    atomicAdd(asum + (size_t)d * N_HEADS + h, a);
}

// ---------------------------------------------------------------------------
// Kernel 6: agg[dst] += a * feat_proj[src]; one wave per edge, float4 lanes
// ---------------------------------------------------------------------------
__global__ __launch_bounds__(256) void gat_aggregate_kernel(
        const float* __restrict__ fp, const float* __restrict__ a,
        const int* __restrict__ src, const int* __restrict__ dst,
        float* __restrict__ agg) {
    const int lane   = threadIdx.x & 31;
    const int eLocal = threadIdx.x >> 5;
    size_t edge = (size_t)blockIdx.x * 8 + eLocal;
    if (edge >= N_EDGES) return;
    int s = src[edge], d = dst[edge];
    int h = lane >> 2;                                  // 4 floats per lane -> head = lane/4
    float aeh = a[edge * N_HEADS + h];
    const float4 v = ((const float4*)(fp + (size_t)s * HD))[lane];
    float* o = agg + (size_t)d * HD + lane * 4;
    atomicAdd(o + 0, v.x * aeh);
    atomicAdd(o + 1, v.y * aeh);
    atomicAdd(o + 2, v.z * aeh);
    atomicAdd(o + 3, v.w * aeh);
}

// ---------------------------------------------------------------------------
// Kernel 7: rst = agg / (attn_sum + 1e-12)  (in place in d_out)
// ---------------------------------------------------------------------------
__global__ void gat_finalize_kernel(float* __restrict__ agg,
                                    const float* __restrict__ asum) {
    size_t i = (size_t)blockIdx.x * blockDim.x + threadIdx.x;
    if (i >= ND) return;
    size_t node = i >> 7;              // /128
    int h = (int)((i >> 4) & 7);       // (i/16)%8
    agg[i] = agg[i] / (asum[node * N_HEADS + h] + 1e-12f);
}

// ---------------------------------------------------------------------------
extern "C" void kernel_launch(void* const* d_in, const int* in_sizes, int n_in,
                              void* d_out, int out_size, void* d_ws, size_t ws_size,
                              hipStream_t stream) {
    const float* feat  = (const float*)d_in[0];   // [N, 128]
    const float* efeat = (const float*)d_in[1];   // [E, 64]
    const float* W     = (const float*)d_in[2];   // [128, 128]
    const float* We    = (const float*)d_in[3];   // [64, 8]
    const int*   src   = (const int*)d_in[4];     // [E]
    const int*   dst   = (const int*)d_in[5];     // [E]

    float* rst_out = (float*)d_out;               // [N, 8, 16] = N*128
    float* a_out   = rst_out + ND;                // [E, 8]

    float* ws        = (float*)d_ws;
    float* feat_proj = ws;                        // N*128 floats
    float* elogit    = feat_proj + ND;            // E*8 floats (becomes ex)
    float* emax      = elogit + EH;               // N*8
    float* denom     = emax + NH;                 // N*8
    float* asum      = denom + NH;                // N*8

    // 0) init accumulators
    gat_init_kernel<<<2048, 256, 0, stream>>>(rst_out, emax, denom, asum);

    // 1) feat_proj = feat @ W  (WMMA f32 16x16x4)
    {
        dim3 blk(32, 8);
        int tiles = N_NODES / 16;                 // 6250
        gat_gemm_node_kernel<<<(tiles + 7) / 8, blk, 0, stream>>>(feat, W, feat_proj);
    }

    // 2) e = e_feat @ W_e  (WMMA f32 16x16x4, N padded in LDS)
    {
        dim3 blk(32, 8);
        int tiles = N_EDGES / 16;                 // 100000
        gat_gemm_edge_kernel<<<tiles / 8, blk, 0, stream>>>(efeat, We, elogit);
    }

    // 3) segment max
    gat_segmax_kernel<<<(unsigned)((EH + 255) / 256), 256, 0, stream>>>(elogit, dst, emax);

    // 4) ex + denom
    gat_exp_kernel<<<(unsigned)((EH + 255) / 256), 256, 0, stream>>>(elogit, dst, emax, denom);

    // 5) a + attn_sum
    gat_attn_kernel<<<(unsigned)((EH + 255) / 256), 256, 0, stream>>>(elogit, dst, denom, a_out, asum);

    // 6) weighted aggregation into d_out (agg region)
    gat_aggregate_kernel<<<N_EDGES / 8, 256, 0, stream>>>(feat_proj, a_out, src, dst, rst_out);

    // 7) finalize rst
    gat_finalize_kernel<<<(unsigned)((ND + 255) / 256), 256, 0, stream>>>(rst_out, asum);
}